// SelfAttention_72121090834619
// MI455X (gfx1250) — compile-verified
//
#include <hip/hip_runtime.h>
#include <math.h>

// ---------------------------------------------------------------------------
// SelfAttention variant on MI455X (gfx1250, wave32).
//   q = x@Wq^T ; k = x@Wk^T ; kkt = (k k^T)*s per head ; dots = (q^T kkt^T)*s
//   out = softplus(dots)  -> (40,32,4096) fp32
// All GEMMs use V_WMMA_F32_16X16X4_F32 (exact fp32, matches reference dtype).
// HBM-bound on the two 67MB weight matrices (~7us floor @ 23.3 TB/s).
// ---------------------------------------------------------------------------

typedef __attribute__((ext_vector_type(2))) float v2f;
typedef __attribute__((ext_vector_type(8))) float v8f;

#define HEADS    8
#define DIM      4096
#define DIM_HEAD 512
#define NB       5
#define NSEQ     32
#define SCALE    0.04419417382415922f   // 512^-0.5

__device__ __forceinline__ float softplusf(float z) {
    // numerically stable log(1+exp(z))
    float r = log1pf(expf(-fabsf(z)));
    return z > 0.0f ? z + r : r;
}

// ---------------------------------------------------------------------------
// Stage 1: q = x@Wq^T (stored transposed as qT[b][e][m]), k = x@Wk^T (natural).
// Tiles: 2 matrices * 10 M-tiles * 256 N-tiles = 5120 wave-tiles, 1 tile/wave.
// Consecutive waves in a block share the same weight columns -> WGP$ reuse.
// Per K-step each lane does one b64 load of x and one b64 load of W.
// ---------------------------------------------------------------------------
__global__ __launch_bounds__(256)
void proj_wmma(const float* __restrict__ x, const float* __restrict__ Wq,
               const float* __restrict__ Wk, float* __restrict__ qT,
               float* __restrict__ kbuf)
{
    const int lane  = threadIdx.x & 31;
    const int wave  = threadIdx.x >> 5;
    const int tile  = blockIdx.x * 8 + wave;      // 0..5119
    const int mat   = tile / 2560;                // 0 = q, 1 = k
    const int rem   = tile - mat * 2560;
    const int ntile = rem / 10;
    const int mtile = rem - ntile * 10;

    const float* __restrict__ W = mat ? Wk : Wq;
    const int half = lane >> 4;                   // K-half select (0 -> K0..1, 1 -> K2..3)
    const int l16  = lane & 15;
    const int m0   = mtile * 16;
    const int n0   = ntile * 16;

    // A: x[m0+l16][k + 2*half .. +1]   B: W[n0+l16][k + 2*half .. +1]
    const float* ap = x + (size_t)(m0 + l16) * DIM + 2 * half;
    const float* bp = W + (size_t)(n0 + l16) * DIM + 2 * half;

    v8f c = {};
    #pragma unroll 8
    for (int k = 0; k < DIM; k += 4) {
        if ((k & 31) == 0)
            __builtin_prefetch(bp + k + 1024, 0, 0);   // global_prefetch_b8, 4KB ahead on W stream
        v2f a = *(const v2f*)(ap + k);
        v2f b = *(const v2f*)(bp + k);
        c = __builtin_amdgcn_wmma_f32_16x16x4_f32(false, a, false, b, (short)0, c, false, false);
    }

    // C/D layout: VGPR v, lane -> row m0 + v + 8*half, col n0 + l16
    const int bb = m0 >> 5;        // batch index (16-row tiles never straddle a batch)
    const int mb = m0 & 31;        // row within batch
    if (mat == 0) {
        // qT[b][e][m] : 8 contiguous floats per lane
        float* o = qT + ((size_t)bb * DIM + n0 + l16) * NSEQ + mb + 8 * half;
        #pragma unroll
        for (int v = 0; v < 8; ++v) o[v] = c[v];
    } else {
        // k natural layout (b, n, 4096)
        #pragma unroll
        for (int v = 0; v < 8; ++v)
            kbuf[(size_t)(bb * NSEQ + mb + 8 * half + v) * DIM + n0 + l16] = c[v];
    }
}

// ---------------------------------------------------------------------------
// Stage 2: kkt[bh] = (k_bh k_bh^T) * SCALE   (40 heads, 32x32 each, K=512)
// 40 * 4 = 160 wave-tiles.
// ---------------------------------------------------------------------------
__global__ __launch_bounds__(256)
void kkt_wmma(const float* __restrict__ kbuf, float* __restrict__ kkt)
{
    const int lane = threadIdx.x & 31;
    const int wave = threadIdx.x >> 5;
    const int tile = blockIdx.x * 8 + wave;       // 0..159
    const int bh = tile >> 2;
    const int it = (tile >> 1) & 1;
    const int jt = tile & 1;
    const int bb = bh >> 3, h = bh & 7;

    const int half = lane >> 4, l16 = lane & 15;
    const float* base = kbuf + (size_t)bb * NSEQ * DIM + h * DIM_HEAD;
    const float* ap = base + (size_t)(it * 16 + l16) * DIM + 2 * half;
    const float* bp = base + (size_t)(jt * 16 + l16) * DIM + 2 * half;

    v8f c = {};
    #pragma unroll 8
    for (int d = 0; d < DIM_HEAD; d += 4) {
        v2f a = *(const v2f*)(ap + d);
        v2f b = *(const v2f*)(bp + d);
        c = __builtin_amdgcn_wmma_f32_16x16x4_f32(false, a, false, b, (short)0, c, false, false);
    }
    #pragma unroll
    for (int v = 0; v < 8; ++v)
        kkt[((size_t)bh * NSEQ + it * 16 + v + 8 * half) * NSEQ + jt * 16 + l16] = c[v] * SCALE;
}

// ---------------------------------------------------------------------------
// Stage 3: dots[bh][i][j] = (sum_m qT[b][i][m] * kkt[bh][j][m]) * SCALE
//          out = softplus(dots), flat layout == reshape(40,32,4096) view.
// Each wave computes a 16x32 strip (both j-tiles) reusing the A fragment.
// 40 * 256 = 10240 wave-tiles.
// ---------------------------------------------------------------------------
__global__ __launch_bounds__(256)
void dots_softplus_wmma(const float* __restrict__ qT, const float* __restrict__ kkt,
                        float* __restrict__ out)
{
    const int lane  = threadIdx.x & 31;
    const int wave  = threadIdx.x >> 5;
    const int tile  = blockIdx.x * 8 + wave;      // 0..10239
    const int bh    = tile >> 8;
    const int itile = tile & 255;
    const int bb    = bh >> 3;
    const int half  = lane >> 4, l16 = lane & 15;

    const float* ap  = qT  + ((size_t)bb * DIM + itile * 16 + l16) * NSEQ + 2 * half;
    const float* bp0 = kkt + ((size_t)bh * NSEQ + l16) * NSEQ + 2 * half;   // j = l16
    const float* bp1 = bp0 + 16 * NSEQ;                                     // j = 16 + l16

    v8f c0 = {}, c1 = {};
    #pragma unroll
    for (int m = 0; m < NSEQ; m += 4) {
        v2f a  = *(const v2f*)(ap  + m);
        v2f b0 = *(const v2f*)(bp0 + m);
        v2f b1 = *(const v2f*)(bp1 + m);
        c0 = __builtin_amdgcn_wmma_f32_16x16x4_f32(false, a, false, b0, (short)0, c0, false, false);
        c1 = __builtin_amdgcn_wmma_f32_16x16x4_f32(false, a, false, b1, (short)0, c1, false, false);
    }

    const size_t rowbase = (size_t)bh * DIM;
    #pragma unroll
    for (int v = 0; v < 8; ++v) {
        const size_t i = (size_t)itile * 16 + v + 8 * half;
        float* o = out + (rowbase + i) * NSEQ;
        o[l16]      = softplusf(c0[v] * SCALE);
        o[16 + l16] = softplusf(c1[v] * SCALE);
    }
}

// ---------------------------------------------------------------------------
extern "C" void kernel_launch(void* const* d_in, const int* in_sizes, int n_in,
                              void* d_out, int out_size, void* d_ws, size_t ws_size,
                              hipStream_t stream)
{
    const float* x  = (const float*)d_in[0];
    const float* Wq = (const float*)d_in[1];
    const float* Wk = (const float*)d_in[2];
    float* out = (float*)d_out;

    // workspace layout (floats): qT | k | kkt   (~5.4 MB total)
    float* qT   = (float*)d_ws;                              // 5*4096*32
    float* kbuf = qT   + (size_t)NB * DIM * NSEQ;            // 5*32*4096
    float* kkt  = kbuf + (size_t)NB * NSEQ * DIM;            // 40*32*32

    proj_wmma         <<< 640, 256, 0, stream>>>(x, Wq, Wk, qT, kbuf);
    kkt_wmma          <<<  20, 256, 0, stream>>>(kbuf, kkt);
    dots_softplus_wmma<<<1280, 256, 0, stream>>>(qT, kkt, out);
}